// CapsuleLayer_77309411361
// MI455X (gfx1250) — compile-verified
//
#include <hip/hip_runtime.h>

// CapsuleLayer dynamic routing, fully fused for MI455X (gfx1250, wave32).
// u_hat recomputed per routing iteration with V_WMMA_F32_16X16X4_F32;
// W (37.7MB) stays L2-resident, so HBM traffic is ~10MB vs 1.2GB for the
// materialized-u_hat approach. W B-operands are software-pipelined one
// ci-iteration ahead so L2 latency hides behind the routing-phase VALU work.

typedef __attribute__((ext_vector_type(2))) float v2f;
typedef __attribute__((ext_vector_type(8))) float v8f;

#define B_TOT 64
#define IN_CAP 2304
#define IDV 8
#define NC 32
#define DV 16
#define NCOL 512            // NC*DV
#define BT 16               // batch tile (WMMA M)
#define CI 48               // input capsules per block
#define NCH (IN_CAP / CI)   // 48 chunks
#define THREADS 512         // 16 waves

// LDS layouts (float indices), padded for conflict-free access
#define IN_K_STRIDE 17
#define IN_CI_STRIDE (IDV * IN_K_STRIDE)   // 136
#define UH_B_STRIDE 37
#define UH_D_STRIDE 587
#define LDS_IN_FLOATS (CI * IN_CI_STRIDE)  // 6528
#define LDS_UH_FLOATS (16 * UH_D_STRIDE)   // 9392
// total LDS = (6528 + 9392) * 4 = 63680 bytes

__global__ __launch_bounds__(THREADS)
void routing_pass(const float* __restrict__ x,     // [64,2304,8]
                  const float* __restrict__ W,     // [2304,32,8,16]
                  const float* __restrict__ bias,  // [1,2304,32,1]
                  const float* __restrict__ veff,  // [64,512] running sum of v
                  float* __restrict__ s_part,      // [NCH,64,512] partials
                  int use_veff)
{
    __shared__ float lds[LDS_IN_FLOATS + LDS_UH_FLOATS];
    float* lin = lds;
    float* luh = lds + LDS_IN_FLOATS;

    const int tid  = threadIdx.x;
    const int lane = tid & 31;
    const int wave = tid >> 5;      // 0..15
    const int ch   = blockIdx.x;    // i-chunk
    const int bt   = blockIdx.y;    // batch tile
    const int b0   = bt * BT;
    const int i0   = ch * CI;

    // ---- stage inputs tile [CI][8k][16b] (padded) into LDS ----
    for (int p = tid; p < CI * BT; p += THREADS) {
        const int ci = p >> 4;
        const int b  = p & 15;
        const float4* g = reinterpret_cast<const float4*>(
            x + ((size_t)(b0 + b) * IN_CAP + (i0 + ci)) * IDV);
        const float4 lo = g[0];
        const float4 hi = g[1];
        float* dst = lin + ci * IN_CI_STRIDE + b;
        dst[0 * IN_K_STRIDE] = lo.x; dst[1 * IN_K_STRIDE] = lo.y;
        dst[2 * IN_K_STRIDE] = lo.z; dst[3 * IN_K_STRIDE] = lo.w;
        dst[4 * IN_K_STRIDE] = hi.x; dst[5 * IN_K_STRIDE] = hi.y;
        dst[6 * IN_K_STRIDE] = hi.z; dst[7 * IN_K_STRIDE] = hi.w;
    }

    // phase-4 role: one local batch per wave, one routing class j per lane
    const int bloc = wave;
    const int j    = lane;

    float vf[DV];
    if (use_veff) {
        const float* gv = veff + (size_t)(b0 + bloc) * NCOL + j * DV;
        #pragma unroll
        for (int d = 0; d < DV; ++d) vf[d] = gv[d];
    } else {
        #pragma unroll
        for (int d = 0; d < DV; ++d) vf[d] = 0.0f;
    }

    float sacc[DV];
    #pragma unroll
    for (int d = 0; d < DV; ++d) sacc[d] = 0.0f;

    // WMMA lane indexing (16x16x4 f32):
    // A 16x4: vgpr0 = K=kb, vgpr1 = K=kb+1; lanes 0-15 -> M=lane (kb=0),
    // lanes 16-31 -> M=lane-16 (kb=2). B 4x16 analogous with cols in lanes.
    const int half = lane >> 4;
    const int arow = lane & 15;
    const int kb   = half * 2;
    const int t0   = wave * 2;   // each wave owns tiles (j) t0, t0+1
    const int dcol = arow;
    const int mbase = half * 8;

    // ---- software-pipelined W B-operand fetch (8 floats/lane/iter) ----
    // wreg[tt] = {B0.x, B0.y, B1.x, B1.y} for tile t0+tt
    float wcur[2][4], wnext[2][4];
    {
        const size_t wrow = (size_t)i0 * NC * (IDV * DV);
        #pragma unroll
        for (int tt = 0; tt < 2; ++tt) {
            const float* wb = W + wrow + (size_t)(t0 + tt) * (IDV * DV) + dcol;
            wcur[tt][0] = wb[(kb + 0) * DV];
            wcur[tt][1] = wb[(kb + 1) * DV];
            wcur[tt][2] = wb[(kb + 4) * DV];
            wcur[tt][3] = wb[(kb + 5) * DV];
        }
    }

    __syncthreads();

    #pragma unroll 2
    for (int ci = 0; ci < CI; ++ci) {
        // ---- A operand from LDS (identical across waves) ----
        const float* abase = lin + ci * IN_CI_STRIDE + arow;
        v2f a0, a1;
        a0.x = abase[(kb + 0) * IN_K_STRIDE];
        a0.y = abase[(kb + 1) * IN_K_STRIDE];
        a1.x = abase[(kb + 4) * IN_K_STRIDE];
        a1.y = abase[(kb + 5) * IN_K_STRIDE];

        // ---- two 16x16 u_hat tiles per wave via f32 WMMA ----
        #pragma unroll
        for (int tt = 0; tt < 2; ++tt) {
            const int t = t0 + tt;                 // routing class j of tile
            v2f bv0, bv1;
            bv0.x = wcur[tt][0];
            bv0.y = wcur[tt][1];
            bv1.x = wcur[tt][2];
            bv1.y = wcur[tt][3];

            v8f acc = {};
            acc = __builtin_amdgcn_wmma_f32_16x16x4_f32(
                false, a0, false, bv0, (short)0, acc, false, false);
            acc = __builtin_amdgcn_wmma_f32_16x16x4_f32(
                false, a1, false, bv1, (short)0, acc, false, false);

            // D: vgpr r -> M=r (lanes 0-15) / M=r+8 (lanes 16-31), col=lane&15
            float* ub = luh + dcol * UH_D_STRIDE + t;
            #pragma unroll
            for (int r = 0; r < 8; ++r)
                ub[(mbase + r) * UH_B_STRIDE] = acc[r];
        }

        // ---- prefetch next iteration's W while routing phase runs ----
        {
            const int cin = (ci + 1 < CI) ? (ci + 1) : ci;   // harmless re-read on last
            const size_t wrow = (size_t)(i0 + cin) * NC * (IDV * DV);
            #pragma unroll
            for (int tt = 0; tt < 2; ++tt) {
                const float* wb = W + wrow + (size_t)(t0 + tt) * (IDV * DV) + dcol;
                wnext[tt][0] = wb[(kb + 0) * DV];
                wnext[tt][1] = wb[(kb + 1) * DV];
                wnext[tt][2] = wb[(kb + 4) * DV];
                wnext[tt][3] = wb[(kb + 5) * DV];
            }
        }

        __syncthreads();

        // ---- routing math: logits, softmax over j (one wave = one b), accum s ----
        float u[DV];
        const float* ur = luh + bloc * UH_B_STRIDE + j;
        #pragma unroll
        for (int d = 0; d < DV; ++d) u[d] = ur[d * UH_D_STRIDE];

        float tlog = bias[(i0 + ci) * NC + j];
        if (use_veff) {
            #pragma unroll
            for (int d = 0; d < DV; ++d) tlog += u[d] * vf[d];
        }
        float m = tlog;
        #pragma unroll
        for (int off = 16; off >= 1; off >>= 1)
            m = fmaxf(m, __shfl_xor(m, off, 32));
        const float e = __expf(tlog - m);
        float ssum = e;
        #pragma unroll
        for (int off = 16; off >= 1; off >>= 1)
            ssum += __shfl_xor(ssum, off, 32);
        const float c = e / ssum;

        #pragma unroll
        for (int d = 0; d < DV; ++d) sacc[d] += c * u[d];

        __syncthreads();

        // rotate pipeline registers (eliminated by the unroll-by-2)
        #pragma unroll
        for (int tt = 0; tt < 2; ++tt)
            #pragma unroll
            for (int q = 0; q < 4; ++q)
                wcur[tt][q] = wnext[tt][q];
    }

    // ---- write deterministic partial sums ----
    float* sp = s_part + ((size_t)ch * B_TOT + b0 + bloc) * NCOL + j * DV;
    #pragma unroll
    for (int d = 0; d < DV; ++d) sp[d] = sacc[d];
}

__global__ void squash_update(const float* __restrict__ s_part,
                              float* __restrict__ veff,
                              float* __restrict__ out,
                              int final_iter)
{
    const int t = blockIdx.x * blockDim.x + threadIdx.x;
    if (t >= B_TOT * NC) return;
    const int b = t >> 5;
    const int j = t & 31;

    float s[DV];
    #pragma unroll
    for (int d = 0; d < DV; ++d) s[d] = 0.0f;
    for (int ch = 0; ch < NCH; ++ch) {     // fixed order -> deterministic
        const float* p = s_part + ((size_t)ch * B_TOT + b) * NCOL + j * DV;
        #pragma unroll
        for (int d = 0; d < DV; ++d) s[d] += p[d];
    }
    float s2 = 0.0f;
    #pragma unroll
    for (int d = 0; d < DV; ++d) s2 += s[d] * s[d];
    const float scale = s2 / ((1.0f + s2) * sqrtf(s2 + 1e-7f));

    if (final_iter) {
        float* dst = out + (size_t)b * NCOL + j * DV;
        #pragma unroll
        for (int d = 0; d < DV; ++d) dst[d] = scale * s[d];
    } else {
        float* dst = veff + (size_t)b * NCOL + j * DV;
        #pragma unroll
        for (int d = 0; d < DV; ++d) dst[d] += scale * s[d];  // veff = v0 (+ v1)
    }
}

__global__ void zero_kernel(float* __restrict__ p, int n)
{
    const int i = blockIdx.x * blockDim.x + threadIdx.x;
    if (i < n) p[i] = 0.0f;
}

extern "C" void kernel_launch(void* const* d_in, const int* in_sizes, int n_in,
                              void* d_out, int out_size, void* d_ws, size_t ws_size,
                              hipStream_t stream)
{
    const float* x    = (const float*)d_in[0];   // [64,2304,8]
    const float* W    = (const float*)d_in[1];   // [2304,32,8,16]
    const float* bias = (const float*)d_in[2];   // [1,2304,32,1]
    float* out = (float*)d_out;                  // [64,32,16]

    float* s_part = (float*)d_ws;                        // NCH*64*512 floats
    float* veff   = s_part + (size_t)NCH * B_TOT * NCOL; // 64*512 floats

    zero_kernel<<<(B_TOT * NCOL + 255) / 256, 256, 0, stream>>>(veff, B_TOT * NCOL);

    for (int r = 0; r < 3; ++r) {
        routing_pass<<<dim3(NCH, B_TOT / BT), THREADS, 0, stream>>>(
            x, W, bias, veff, s_part, r > 0 ? 1 : 0);
        squash_update<<<(B_TOT * NC + 255) / 256, 256, 0, stream>>>(
            s_part, veff, out, r == 2 ? 1 : 0);
    }
}